// TricksComb_5944234737800
// MI455X (gfx1250) — compile-verified
//
#include <hip/hip_runtime.h>
#include <hip/hip_bf16.h>

typedef __attribute__((ext_vector_type(2))) float v2f;
typedef __attribute__((ext_vector_type(8))) float v8f;

#define N_NODES 50000

// ---------------------------------------------------------------------------
// Degree accumulation: deg_out[src[e]]++, deg_in[dst[e]]++
// ---------------------------------------------------------------------------
__global__ void degree_kernel(const int* __restrict__ src,
                              const int* __restrict__ dst,
                              int* __restrict__ degO,
                              int* __restrict__ degI, int nE) {
    int e = blockIdx.x * blockDim.x + threadIdx.x;
    if (e < nE) {
        atomicAdd(&degO[src[e]], 1);
        atomicAdd(&degI[dst[e]], 1);
    }
}

// norm[i] = 1/sqrt(max(deg[i],1))  (matches clip(deg,1)^-0.5)
__global__ void norm_kernel(const int* __restrict__ deg,
                            float* __restrict__ norm, int n) {
    int i = blockIdx.x * blockDim.x + threadIdx.x;
    if (i < n) {
        float d = (float)(deg[i] < 1 ? 1 : deg[i]);
        norm[i] = 1.0f / sqrtf(d);
    }
}

// ---------------------------------------------------------------------------
// WMMA f32 GEMM: P[M,N] = (A[M,K] * scale[M]) @ W[K,N]
// One wave per 16x16 tile; K-loop in steps of 4 via V_WMMA_F32_16X16X4_F32.
// A layout  (16x4  f32): lane = hi*16 + m (m = lane&15), a[v] = A[m][k0+2*hi+v]
// B layout  (4x16  f32): lane = hi*16 + n,               b[v] = W[k0+2*hi+v][n]
// C/D layout(16x16 f32): VGPR r, lane L -> (M = r + 8*(L>>4), N = L&15)
// ---------------------------------------------------------------------------
template <int K, int N>
__global__ __launch_bounds__(256) void gemm_wmma(
    const float* __restrict__ A, const float* __restrict__ scale,
    const float* __restrict__ W, float* __restrict__ P, int M) {
    const int wave = threadIdx.x >> 5;
    const int lane = threadIdx.x & 31;
    const int tilesN = N / 16;
    const int tile = blockIdx.x * 8 + wave;
    if (tile >= (M / 16) * tilesN) return;  // wave-uniform guard (EXEC all-1 for WMMA)

    const int mb = tile / tilesN;
    const int nb = tile - mb * tilesN;
    const int m0 = mb * 16;
    const int n0 = nb * 16;
    const int ln = lane & 15;  // m for A, n for B/D columns
    const int hi = lane >> 4;  // selects K-pair (A/B) and M-half (D)

    const float sm = scale[m0 + ln];                       // fused left-norm
    const float* Arow = A + (size_t)(m0 + ln) * K + 2 * hi;
    const float* Wp   = W + n0 + ln;

    v8f c = {0.f, 0.f, 0.f, 0.f, 0.f, 0.f, 0.f, 0.f};
#pragma unroll 8
    for (int k0 = 0; k0 < K; k0 += 4) {
        v2f a = *(const v2f*)(Arow + k0);
        a.x *= sm;
        a.y *= sm;
        v2f b;
        b.x = Wp[(size_t)(k0 + 2 * hi) * N];
        b.y = Wp[(size_t)(k0 + 2 * hi + 1) * N];
        // 8 args: (neg_a, A, neg_b, B, c_mod, C, reuse_a, reuse_b)
        c = __builtin_amdgcn_wmma_f32_16x16x4_f32(
            false, a, false, b, (short)0, c, false, false);
    }

    float* Pp = P + (size_t)(m0 + 8 * hi) * N + n0 + ln;
#pragma unroll
    for (int r = 0; r < 8; ++r) Pp[(size_t)r * N] = c[r];
}

// ---------------------------------------------------------------------------
// Edge scatter: G[dst[e]][:] += P[src[e]][:]   (one wave per edge)
// ---------------------------------------------------------------------------
template <int F>
__global__ __launch_bounds__(256) void scatter_kernel(
    const float* __restrict__ P, const int* __restrict__ src,
    const int* __restrict__ dst, float* __restrict__ G, int nE) {
    int e = blockIdx.x * 8 + (threadIdx.x >> 5);
    if (e >= nE) return;
    const int lane = threadIdx.x & 31;
    constexpr int PER = F / 32;  // 4 (F=128) or 2 (F=64)
    const int s = src[e];
    const int d = dst[e];
    const float* hp = P + (size_t)s * F + lane * PER;
    float* gp = G + (size_t)d * F + lane * PER;
    float v[PER];
#pragma unroll
    for (int j = 0; j < PER; ++j) v[j] = hp[j];
#pragma unroll
    for (int j = 0; j < PER; ++j)
        __hip_atomic_fetch_add(gp + j, v[j], __ATOMIC_RELAXED,
                               __HIP_MEMORY_SCOPE_AGENT);
}

// ---------------------------------------------------------------------------
// out[i,j] = (relu?)(G[i,j] * in_norm[i] + b[j])
// ---------------------------------------------------------------------------
template <int F, bool RELU>
__global__ void pointwise_kernel(const float* __restrict__ G,
                                 const float* __restrict__ inorm,
                                 const float* __restrict__ bias,
                                 float* __restrict__ O, int total) {
    int idx = blockIdx.x * blockDim.x + threadIdx.x;
    if (idx >= total) return;
    int i = idx / F;
    int j = idx - i * F;
    float v = G[idx] * inorm[i] + bias[j];
    if (RELU) v = fmaxf(v, 0.0f);
    O[idx] = v;
}

// ---------------------------------------------------------------------------
extern "C" void kernel_launch(void* const* d_in, const int* in_sizes, int n_in,
                              void* d_out, int out_size, void* d_ws,
                              size_t ws_size, hipStream_t stream) {
    const float* x  = (const float*)d_in[0];
    const int*   ei = (const int*)d_in[1];  // [2, nE] flattened
    const float* W0 = (const float*)d_in[2];
    const float* b0 = (const float*)d_in[3];
    const float* W1 = (const float*)d_in[4];
    const float* b1 = (const float*)d_in[5];
    const float* W2 = (const float*)d_in[6];
    const float* b2 = (const float*)d_in[7];
    const float* W3 = (const float*)d_in[8];
    const float* b3 = (const float*)d_in[9];

    const int nN = N_NODES;
    const int nE = in_sizes[1] / 2;
    const int* src = ei;
    const int* dst = ei + nE;

    // Carve workspace
    char* ws = (char*)d_ws;
    size_t off = 0;
    auto alloc = [&](size_t bytes) -> void* {
        off = (off + 255) & ~(size_t)255;
        void* p = ws + off;
        off += bytes;
        return p;
    };
    int*   degO  = (int*)alloc((size_t)nN * sizeof(int));
    int*   degI  = (int*)alloc((size_t)nN * sizeof(int));
    float* onorm = (float*)alloc((size_t)nN * sizeof(float));
    float* inorm = (float*)alloc((size_t)nN * sizeof(float));
    float* P = (float*)alloc((size_t)nN * 128 * sizeof(float));  // pre-agg
    float* G = (float*)alloc((size_t)nN * 128 * sizeof(float));  // agg
    float* H = (float*)alloc((size_t)nN * 128 * sizeof(float));  // layer input

    // --- degree norms -------------------------------------------------------
    hipMemsetAsync(degO, 0, (size_t)nN * sizeof(int), stream);
    hipMemsetAsync(degI, 0, (size_t)nN * sizeof(int), stream);
    degree_kernel<<<(nE + 255) / 256, 256, 0, stream>>>(src, dst, degO, degI, nE);
    norm_kernel<<<(nN + 255) / 256, 256, 0, stream>>>(degO, onorm, nN);
    norm_kernel<<<(nN + 255) / 256, 256, 0, stream>>>(degI, inorm, nN);

    const int tilesM = nN / 16;  // 3125, exact

    // --- layer 0: 256 -> 128, relu -----------------------------------------
    gemm_wmma<256, 128><<<(tilesM * 8 + 7) / 8, 256, 0, stream>>>(x, onorm, W0, P, nN);
    hipMemsetAsync(G, 0, (size_t)nN * 128 * sizeof(float), stream);
    scatter_kernel<128><<<(nE + 7) / 8, 256, 0, stream>>>(P, src, dst, G, nE);
    pointwise_kernel<128, true><<<(nN * 128 + 255) / 256, 256, 0, stream>>>(
        G, inorm, b0, H, nN * 128);

    // --- layer 1: 128 -> 128, relu -----------------------------------------
    gemm_wmma<128, 128><<<(tilesM * 8 + 7) / 8, 256, 0, stream>>>(H, onorm, W1, P, nN);
    hipMemsetAsync(G, 0, (size_t)nN * 128 * sizeof(float), stream);
    scatter_kernel<128><<<(nE + 7) / 8, 256, 0, stream>>>(P, src, dst, G, nE);
    pointwise_kernel<128, true><<<(nN * 128 + 255) / 256, 256, 0, stream>>>(
        G, inorm, b1, H, nN * 128);

    // --- layer 2: 128 -> 128, relu -----------------------------------------
    gemm_wmma<128, 128><<<(tilesM * 8 + 7) / 8, 256, 0, stream>>>(H, onorm, W2, P, nN);
    hipMemsetAsync(G, 0, (size_t)nN * 128 * sizeof(float), stream);
    scatter_kernel<128><<<(nE + 7) / 8, 256, 0, stream>>>(P, src, dst, G, nE);
    pointwise_kernel<128, true><<<(nN * 128 + 255) / 256, 256, 0, stream>>>(
        G, inorm, b2, H, nN * 128);

    // --- layer 3: 128 -> 64, no relu, write d_out --------------------------
    gemm_wmma<128, 64><<<(tilesM * 4 + 7) / 8, 256, 0, stream>>>(H, onorm, W3, P, nN);
    hipMemsetAsync(G, 0, (size_t)nN * 64 * sizeof(float), stream);
    scatter_kernel<64><<<(nE + 7) / 8, 256, 0, stream>>>(P, src, dst, G, nE);
    pointwise_kernel<64, false><<<(nN * 64 + 255) / 256, 256, 0, stream>>>(
        G, inorm, b3, (float*)d_out, nN * 64);

    (void)n_in; (void)out_size; (void)ws_size;
}